// FactorPredictor_65274912965359
// MI455X (gfx1250) — compile-verified
//
#include <hip/hip_runtime.h>
#include <hip/hip_bf16.h>
#include <math.h>

// Problem constants (from the reference)
#define NS_   100000
#define H_    128
#define K_    4

// Launch geometry
#define NBLK  256            // blocks for the streaming pass
#define TPB   256            // 8 wave32 per block
#define WPB   (TPB / 32)     // waves per block
#define TW    (NBLK * WPB)   // total waves
#define WS_STRIDE 260        // per-block ws: 128 hpos + 128 hneg + spos + sneg (+2 pad)

typedef __attribute__((ext_vector_type(2))) float v2f;
typedef __attribute__((ext_vector_type(8))) float v8f;

// ---------------------------------------------------------------------------
// Pass 1: single streaming pass over e (51.2 MB, the roofline term).
// One wave32 processes one row per iteration: lane l owns h = 4l..4l+3
// (one global_load_b128 per lane; 32 lanes = 512B = the full row, perfectly
// coalesced).  Cross-lane shfl_xor reduces q·e and e·e, then each lane
// accumulates relu(+/-cos)*e into registers.  Per-block partials are reduced
// deterministically through LDS and written to ws (no float atomics ->
// bit-deterministic across graph replays).
// ---------------------------------------------------------------------------
__global__ __launch_bounds__(TPB) void fp_pass1(const float* __restrict__ e,
                                                const float* __restrict__ q,
                                                float* __restrict__ ws) {
    __shared__ float sh_hp[WPB][H_];
    __shared__ float sh_hn[WPB][H_];
    __shared__ float sh_sp[WPB];
    __shared__ float sh_sn[WPB];

    const int lane = threadIdx.x & 31;
    const int wave = threadIdx.x >> 5;
    const int gw   = blockIdx.x * WPB + wave;

    // ||q|| (cheap, per wave; q is L2/L0 resident)
    const float4 q4 = reinterpret_cast<const float4*>(q)[lane];
    float qq = q4.x * q4.x + q4.y * q4.y + q4.z * q4.z + q4.w * q4.w;
#pragma unroll
    for (int off = 16; off > 0; off >>= 1) qq += __shfl_xor(qq, off, 32);
    const float qn = sqrtf(qq);

    float4 accp = {0.f, 0.f, 0.f, 0.f};
    float4 accn = {0.f, 0.f, 0.f, 0.f};
    float  sp = 0.f, sn = 0.f;

    for (int n = gw; n < NS_; n += TW) {
        const float4 e4 =
            reinterpret_cast<const float4*>(e + (size_t)n * H_)[lane];
        float pd = q4.x * e4.x + q4.y * e4.y + q4.z * e4.z + q4.w * e4.w;
        float ps = e4.x * e4.x + e4.y * e4.y + e4.z * e4.z + e4.w * e4.w;
#pragma unroll
        for (int off = 16; off > 0; off >>= 1) {
            pd += __shfl_xor(pd, off, 32);
            ps += __shfl_xor(ps, off, 32);
        }
        const float c  = pd / (qn * sqrtf(ps));   // cosine(q, e_n)
        const float ap = fmaxf(c, 0.f);
        const float an = fmaxf(-c, 0.f);
        sp += ap;                                  // identical in all lanes
        sn += an;
        accp.x += ap * e4.x; accp.y += ap * e4.y;
        accp.z += ap * e4.z; accp.w += ap * e4.w;
        accn.x += an * e4.x; accn.y += an * e4.y;
        accn.z += an * e4.z; accn.w += an * e4.w;
    }

    sh_hp[wave][lane * 4 + 0] = accp.x;
    sh_hp[wave][lane * 4 + 1] = accp.y;
    sh_hp[wave][lane * 4 + 2] = accp.z;
    sh_hp[wave][lane * 4 + 3] = accp.w;
    sh_hn[wave][lane * 4 + 0] = accn.x;
    sh_hn[wave][lane * 4 + 1] = accn.y;
    sh_hn[wave][lane * 4 + 2] = accn.z;
    sh_hn[wave][lane * 4 + 3] = accn.w;
    if (lane == 0) { sh_sp[wave] = sp; sh_sn[wave] = sn; }
    __syncthreads();

    float* wsb = ws + (size_t)blockIdx.x * WS_STRIDE;
    const int t = threadIdx.x;
    if (t < H_) {
        float hp = 0.f, hn = 0.f;
#pragma unroll
        for (int w = 0; w < WPB; ++w) { hp += sh_hp[w][t]; hn += sh_hn[w][t]; }
        wsb[t]      = hp;
        wsb[H_ + t] = hn;
    } else if (t == H_) {
        float s = 0.f;
#pragma unroll
        for (int w = 0; w < WPB; ++w) s += sh_sp[w];
        wsb[2 * H_] = s;
    } else if (t == H_ + 1) {
        float s = 0.f;
#pragma unroll
        for (int w = 0; w < WPB; ++w) s += sh_sn[w];
        wsb[2 * H_ + 1] = s;
    }
}

// ---------------------------------------------------------------------------
// Pass 2 (tiny): reduce block partials, then the 2x128 @ 128x2 projection
// [hpos;hneg] x [mu_w;sigma_w]^T via 32 chained V_WMMA_F32_16X16X4_F32
// (A 16x4: lanes 0-15 hold K=0,1, lanes 16-31 hold K=2,3; D[m,n] for
// m,n in {0,1} lands in acc VGPR m at lane n).
//
// Fragment build is branch-free: all operands staged in LDS, every lane
// loads unconditionally and selects its matrix row via 0/1 float masks
// (avoids the exec-mask save/restore per scalar load seen in round 1).
// ---------------------------------------------------------------------------
__global__ __launch_bounds__(128) void fp_finalize(
    const float* __restrict__ ws,
    const float* __restrict__ w_key,  const float* __restrict__ w_value,
    const float* __restrict__ mu_w,   const float* __restrict__ mu_b,
    const float* __restrict__ sg_w,   const float* __restrict__ sg_b,
    float* __restrict__ out) {
    __shared__ float sh_hp[H_];
    __shared__ float sh_hn[H_];
    __shared__ float sh_mu[H_];
    __shared__ float sh_sg[H_];
    __shared__ float sh_s[2];

    const int t = threadIdx.x;
    if (t < H_) {
        float hp = 0.f, hn = 0.f;
        for (int b = 0; b < NBLK; ++b) {
            hp += ws[(size_t)b * WS_STRIDE + t];
            hn += ws[(size_t)b * WS_STRIDE + H_ + t];
        }
        sh_hp[t] = hp;
        sh_hn[t] = hn;
        sh_mu[t] = mu_w[t];     // stage projection weights in LDS once
        sh_sg[t] = sg_w[t];
    }
    if (t < 2) {
        float s = 0.f;
        for (int b = 0; b < NBLK; ++b) s += ws[(size_t)b * WS_STRIDE + 2 * H_ + t];
        sh_s[t] = s;
    }
    __syncthreads();

    if (t < 32) {               // wave 0 only; EXEC all-1s within the wave
        const int   lane = t;
        const int   row  = lane & 15;          // A: M index / B: N index
        const int   k0   = (lane >> 4) * 2;    // K pair handled by this lane half
        const float m0   = (row == 0) ? 1.f : 0.f;   // select matrix row 0
        const float m1   = (row == 1) ? 1.f : 0.f;   // select matrix row 1
        v8f acc = {0.f, 0.f, 0.f, 0.f, 0.f, 0.f, 0.f, 0.f};
#pragma unroll
        for (int h0 = 0; h0 < H_; h0 += 4) {
            v2f a, b;
            a[0] = m0 * sh_hp[h0 + k0]     + m1 * sh_hn[h0 + k0];
            a[1] = m0 * sh_hp[h0 + k0 + 1] + m1 * sh_hn[h0 + k0 + 1];
            b[0] = m0 * sh_mu[h0 + k0]     + m1 * sh_sg[h0 + k0];
            b[1] = m0 * sh_mu[h0 + k0 + 1] + m1 * sh_sg[h0 + k0 + 1];
            acc = __builtin_amdgcn_wmma_f32_16x16x4_f32(
                false, a, false, b, (short)0, acc, false, false);
        }
        // D[0,0]=hpos·mu  D[0,1]=hpos·sg  D[1,0]=hneg·mu  D[1,1]=hneg·sg
        const float v00 = __shfl(acc[0], 0, 32);
        const float v01 = __shfl(acc[0], 1, 32);
        const float v10 = __shfl(acc[1], 0, 32);
        const float v11 = __shfl(acc[1], 1, 32);

        if (lane < K_) {
            const float spc = fmaxf(sh_s[0], 1e-6f);
            const float snc = fmaxf(sh_s[1], 1e-6f);
            const bool  pos = w_key[lane] > 0.f;   // only sign(w_key) matters
            const float dm  = pos ? v00 / spc : v10 / snc;
            const float dg  = pos ? v01 / spc : v11 / snc;
            const float wv  = w_value[lane];
            out[lane] = wv * dm + mu_b[0];
            const float x = wv * dg + sg_b[0];
            // numerically-stable softplus, matches jax.nn.softplus
            out[K_ + lane] = fmaxf(x, 0.f) + log1pf(expf(-fabsf(x)));
        }
    }
}

extern "C" void kernel_launch(void* const* d_in, const int* in_sizes, int n_in,
                              void* d_out, int out_size, void* d_ws, size_t ws_size,
                              hipStream_t stream) {
    const float* e     = (const float*)d_in[0];
    const float* w_key = (const float*)d_in[1];
    const float* w_val = (const float*)d_in[2];
    const float* q     = (const float*)d_in[3];
    const float* mu_w  = (const float*)d_in[4];
    const float* mu_b  = (const float*)d_in[5];
    const float* sg_w  = (const float*)d_in[6];
    const float* sg_b  = (const float*)d_in[7];
    float* out = (float*)d_out;
    float* ws  = (float*)d_ws;   // needs NBLK*WS_STRIDE*4 = 266,240 bytes

    fp_pass1<<<NBLK, TPB, 0, stream>>>(e, q, ws);
    fp_finalize<<<1, 128, 0, stream>>>(ws, w_key, w_val, mu_w, mu_b, sg_w, sg_b, out);
}